// EsinerAgent_37812892074301
// MI455X (gfx1250) — compile-verified
//
#include <hip/hip_runtime.h>

// Eisner DP chart fill for MI455X (gfx1250).
// One workgroup (256 threads = 8 wave32) per batch element.
// Whole score chart (4 planes, ~257KB) lives in static LDS — CDNA5 allows up
// to 320KB per workgroup (CDNA5 ISA); this also makes hipcc verify
// the allocation against the gfx1250 group-segment limit at compile time.
// Final chart is written back with gfx1250 async LDS->global stores
// (GLOBAL_STORE_ASYNC_FROM_LDS_B32, tracked by ASYNCcnt; CDNA5 ISA).

#define NNODES 128
#define S_PAD  129            // pad so column reads hit all 64 LDS banks (gcd(129,64)=1)
#define NEGV   (-9999.0f)
#define THRESHV (-9000.0f)
#define ARC_BONUS 5.0f

#define NT      256           // threads per block (8 wave32)
#define NWAVES  8

__device__ __forceinline__ void argmax_combine(float& v, int& m, float ov, int om) {
    // max value; on exact tie keep the SMALLER split index (jnp.argmax = first occurrence)
    if (ov > v || (ov == v && om < m)) { v = ov; m = om; }
}

__device__ __forceinline__ void wave_argmax(float& v, int& m) {
#pragma unroll
    for (int off = 16; off > 0; off >>= 1) {
        float ov = __shfl_xor(v, off, 32);
        int   om = __shfl_xor(m, off, 32);
        argmax_combine(v, m, ov, om);
    }
}

// CDNA5 async store: LDS -> global, tracked by ASYNCcnt (CDNA5 ISA)
__device__ __forceinline__ void async_store_lds_to_global(float* g, const float* l) {
    unsigned int loff = (unsigned int)(unsigned long long)(const void*)l; // low 32b of generic ptr = LDS byte address
    asm volatile("global_store_async_from_lds_b32 %0, %1, off"
                 :: "v"(g), "v"(loff)
                 : "memory");
}

// Static LDS: 2 padded planes (row+col access) + 2 unpadded planes = 263,168 B.
__shared__ float s11 [NNODES * S_PAD];   // [i][q] row-major        (row + column reads)
__shared__ float s01 [NNODES * S_PAD];   // [i][q] row-major        (row + column reads)
__shared__ float s10 [NNODES * NNODES];  // [i][q] row-major        (row reads only)
__shared__ float s00t[NNODES * NNODES];  // s00t[j*N+i] = S00[i][j] (column-of-S00 = row here)

__global__ __launch_bounds__(NT, 1)
void eisner_fill_chart(const float* __restrict__ vinfo_all,
                       float* __restrict__ scores_out,
                       int* __restrict__ bt_out) {
    constexpr int N = NNODES;

    const int b = blockIdx.x;
    const float* __restrict__ vinfo = vinfo_all + (size_t)b * N * N;
    float* __restrict__ Sg  = scores_out + (size_t)b * N * N * 4;   // [i][j][dir][comp]
    int*   __restrict__ BTg = bt_out     + (size_t)b * N * N * 4;

    const int tid  = threadIdx.x;
    const int lane = tid & 31;
    const int wave = tid >> 5;

    // ---- init: chart = NEG, diagonal = 0; global BT = 0 (re-done every call) ----
    for (int idx = tid; idx < N * S_PAD; idx += NT) { s11[idx] = NEGV; s01[idx] = NEGV; }
    for (int idx = tid; idx < N * N;     idx += NT) { s10[idx] = NEGV; s00t[idx] = NEGV; }
    for (int idx = tid; idx < N * N * 4; idx += NT) BTg[idx] = 0;
    __syncthreads();
    for (int i = tid; i < N; i += NT) {
        s11[i * S_PAD + i] = 0.0f; s01[i * S_PAD + i] = 0.0f;
        s10[i * N + i]     = 0.0f; s00t[i * N + i]    = 0.0f;
    }
    __syncthreads();

    // ---- DP over span width k; all reads at step k touch width < k cells ----
    for (int k = 1; k < N; ++k) {
        const int nspan = N - k;
        for (int i = wave; i < nspan; i += NWAVES) {   // one wave per span (i, j)
            const int j = i + k;
            const float vL = vinfo[j * N + i];
            const float vR = vinfo[i * N + j];

            float b00 = NEGV, b10 = NEGV, b01 = NEGV, b11 = NEGV;
            int   a00 = 0,    a10 = 0,    a01 = 0,    a11 = 0;

            for (int m = lane; m < k; m += 32) {       // split q = i + m
                const int q = i + m;
                const float base = s11[i * S_PAD + q] + s01[(q + 1) * S_PAD + j];
                const float c0 = fmaxf(base + vL + ARC_BONUS, NEGV);   // (0,0) candidate
                const float c1 = fmaxf(base + vR + ARC_BONUS, NEGV);   // (1,0) candidate
                if (c0 > b00) { b00 = c0; a00 = m; }
                if (c1 > b10) { b10 = c1; a10 = m; }
                if (m >= 1) {  // m in [1, k): complete-cell candidates
                    const float cc01 = fmaxf(s01[i * S_PAD + q] + s00t[j * N + q], NEGV);
                    const float cc11 = fmaxf(s10[i * N + q] + s11[q * S_PAD + j], NEGV);
                    if (cc01 > b01) { b01 = cc01; a01 = m; }
                    if (cc11 > b11) { b11 = cc11; a11 = m; }
                }
            }
            wave_argmax(b00, a00);
            wave_argmax(b10, a10);
            wave_argmax(b01, a01);
            wave_argmax(b11, a11);

            if (lane == 0) {
                // part00: (0,0) value at split q=i exactly as the source reads it
                const float part00raw = fmaxf(s01[(i + 1) * S_PAD + j] + vL + ARC_BONUS, NEGV);
                const float part00 = (part00raw > THRESHV) ? part00raw : NEGV;
                // (0,1): m=0 candidate (part00) comes FIRST -> wins ties
                float m01; int q01m;
                if (part00 >= b01) { m01 = part00; q01m = 0; }
                else               { m01 = b01;    q01m = a01; }
                // new10: freshly updated (1,0) cell feeds (1,1) at q = j (index m=k, LAST)
                const bool  ok10  = b10 > THRESHV;
                const float new10 = ok10 ? b10 : NEGV;
                float m11; int q11m;
                if (new10 > b11) { m11 = new10; q11m = k; }
                else             { m11 = b11;   q11m = a11; }

                const bool ok00 = b00 > THRESHV;
                const bool ok01 = m01 > THRESHV;
                const bool ok11 = m11 > THRESHV;

                if (ok00) s00t[j * N + i]    = b00;
                if (ok10) s10[i * N + j]     = b10;
                if (ok01) s01[i * S_PAD + j] = m01;
                if (ok11) s11[i * S_PAD + j] = m11;

                int* bt = BTg + (size_t)(i * N + j) * 4;   // [dir][comp]: 0=(0,0) 1=(0,1) 2=(1,0) 3=(1,1)
                bt[0] = ok00 ? (i + a00)  : 0;
                bt[1] = ok01 ? (i + q01m) : 0;
                bt[2] = ok10 ? (i + a10)  : 0;
                bt[3] = ok11 ? (i + q11m) : 0;
            }
        }
        __syncthreads();   // width-k cells complete before width-(k+1) reads them
    }
    __syncthreads();

    // ---- writeback: async LDS -> global for the whole chart (incl. NEG / diag cells) ----
    for (int idx = tid; idx < N * N; idx += NT) {
        const int i = idx >> 7;
        const int j = idx & (N - 1);
        float* g = Sg + (size_t)idx * 4;
        async_store_lds_to_global(g + 0, &s00t[j * N + i]);     // (0,0)
        async_store_lds_to_global(g + 1, &s01[i * S_PAD + j]);  // (0,1)
        async_store_lds_to_global(g + 2, &s10[i * N + j]);      // (1,0)
        async_store_lds_to_global(g + 3, &s11[i * S_PAD + j]);  // (1,1)
    }
    asm volatile("s_wait_asynccnt 0x0" ::: "memory");
}

extern "C" void kernel_launch(void* const* d_in, const int* in_sizes, int n_in,
                              void* d_out, int out_size, void* d_ws, size_t ws_size,
                              hipStream_t stream) {
    (void)d_ws; (void)ws_size; (void)out_size;
    constexpr int N = NNODES;
    const int B = (n_in > 1) ? in_sizes[1] : 64;   // b_buffer_size has B elements

    const float* vinfo = (const float*)d_in[0];
    float* scores = (float*)d_out;                               // [B,N,N,2,2] f32
    int*   bt     = (int*)(scores + (size_t)B * N * N * 4);      // [B,N,N,2,2] i32

    eisner_fill_chart<<<B, NT, 0, stream>>>(vinfo, scores, bt);
}